// SpectralPooling_14766097563940
// MI455X (gfx1250) — compile-verified
//
#include <hip/hip_runtime.h>
#include <math.h>

typedef __attribute__((ext_vector_type(2))) float v2f;
typedef __attribute__((ext_vector_type(8))) float v8f;

#define DPI 3.14159265358979323846

// ---------------------------------------------------------------------------
// Setup: build Q = (Mi^3)[:, :26] @ (M64^3)[:26, :]  (32 x 64, fp32)
// Matrices are fp32-rounded (matching reference) but products done in fp64.
// Single block, phases separated by threadfence+syncthreads.
// ws layout (doubles): M64[4096] T1[4096] M3[4096] Mi[1024] A1[1024] Mi3[1024]
// ---------------------------------------------------------------------------
__global__ void spool_setup_kernel(double* __restrict__ ws, float* __restrict__ Qf) {
    double* M64 = ws;
    double* T1  = ws + 4096;
    double* M3  = ws + 8192;
    double* Mi  = ws + 12288;  // Mi[i][k] = dct32[k][i]  (32x32)
    double* A1  = ws + 13312;
    double* Mi3 = ws + 14336;
    const int t = threadIdx.x;

    // Phase 0: fill M64 (dct64) and Mi (dct32 transposed), fp32-rounded
    for (int idx = t; idx < 4096; idx += 256) {
        int k = idx >> 6, n = idx & 63;
        double v = (k == 0) ? (1.0 / sqrt(64.0))
                            : sqrt(2.0 / 64.0) * cos(DPI * (2.0 * n + 1.0) * k / 128.0);
        M64[idx] = (double)(float)v;
    }
    for (int idx = t; idx < 1024; idx += 256) {
        int i = idx >> 5, k = idx & 31;
        double v = (k == 0) ? (1.0 / sqrt(32.0))
                            : sqrt(2.0 / 32.0) * cos(DPI * (2.0 * i + 1.0) * k / 64.0);
        Mi[idx] = (double)(float)v;
    }
    __threadfence(); __syncthreads();

    // Phase 1: T1 = M64 * M64
    for (int idx = t; idx < 4096; idx += 256) {
        int i = idx >> 6, j = idx & 63;
        double s = 0.0;
        for (int k = 0; k < 64; ++k) s += M64[i * 64 + k] * M64[k * 64 + j];
        T1[idx] = s;
    }
    __threadfence(); __syncthreads();

    // Phase 2: M3 = T1 * M64
    for (int idx = t; idx < 4096; idx += 256) {
        int i = idx >> 6, j = idx & 63;
        double s = 0.0;
        for (int k = 0; k < 64; ++k) s += T1[i * 64 + k] * M64[k * 64 + j];
        M3[idx] = s;
    }
    __threadfence(); __syncthreads();

    // Phase 3: A1 = Mi * Mi
    for (int idx = t; idx < 1024; idx += 256) {
        int i = idx >> 5, j = idx & 31;
        double s = 0.0;
        for (int k = 0; k < 32; ++k) s += Mi[i * 32 + k] * Mi[k * 32 + j];
        A1[idx] = s;
    }
    __threadfence(); __syncthreads();

    // Phase 4: Mi3 = A1 * Mi
    for (int idx = t; idx < 1024; idx += 256) {
        int i = idx >> 5, j = idx & 31;
        double s = 0.0;
        for (int k = 0; k < 32; ++k) s += A1[i * 32 + k] * Mi[k * 32 + j];
        Mi3[idx] = s;
    }
    __threadfence(); __syncthreads();

    // Phase 5: Q[j][h] = sum_{f<26} Mi3[j][f] * M3[f][h]
    for (int idx = t; idx < 2048; idx += 256) {
        int j = idx >> 6, h = idx & 63;
        double s = 0.0;
        for (int f = 0; f < 26; ++f) s += Mi3[j * 32 + f] * M3[f * 64 + h];
        Qf[idx] = (float)s;
    }
}

// ---------------------------------------------------------------------------
// Main: one (b,c,d) slice per wave.
//   out_slice(32x32) = Q(32x64) @ x_slice(64 x 32-cols, cols>=26 zeroed)
// WMMA f32 16x16x4: 2 M-tiles x 2 N-tiles x 16 k-steps = 64 wmma per slice.
// ---------------------------------------------------------------------------
__global__ void __launch_bounds__(256) spool_main_kernel(
        const float* __restrict__ x, const float* __restrict__ Qf,
        float* __restrict__ out) {
    const int lane = threadIdx.x & 31;
    const int l16  = lane & 15;
    const int half = lane >> 4;                 // 0 or 1
    const int gwave  = blockIdx.x * (blockDim.x >> 5) + (threadIdx.x >> 5);
    const int nwaves = gridDim.x * (blockDim.x >> 5);

    // Preload A fragments of Q (resident across all slices this wave handles).
    // A 16x4 f32 layout: lane holds row m=lane%16; VGPR v holds K = 2*half + v.
    v2f a0[16], a1[16];
#pragma unroll
    for (int ks = 0; ks < 16; ++ks) {
        const int col = ks * 4 + 2 * half;
        a0[ks] = *(const v2f*)(Qf + (l16)      * 64 + col);
        a1[ks] = *(const v2f*)(Qf + (16 + l16) * 64 + col);
    }

    const int NS = 8 * 32 * 32;                 // 8192 output slices
    for (int s = gwave; s < NS; s += nwaves) {  // wave-uniform loop
        const int d  = s & 31;
        const int bc = s >> 5;
        float* __restrict__ os = out + (size_t)s * 1024;

        if (d < 26) {                            // wave-uniform branch
            const float* __restrict__ xs = x + ((size_t)bc * 64 + d) * 4096;
            v8f c00 = {}, c01 = {}, c10 = {}, c11 = {};
#pragma unroll
            for (int ks = 0; ks < 16; ++ks) {
                // B 4x16 f32 layout: lane holds col n=lane%16; VGPR v holds
                // K-row = kb + 2*half + v.
                const int r0 = ks * 4 + 2 * half;
                v2f b0, b1;
                b0.x = xs[r0 * 64 + l16];
                b0.y = xs[r0 * 64 + 64 + l16];
                const float bx = xs[r0 * 64 + 16 + l16];
                const float by = xs[r0 * 64 + 80 + l16];
                const bool zc = (l16 >= 10);     // global col 16+l16 >= 26
                b1.x = zc ? 0.0f : bx;
                b1.y = zc ? 0.0f : by;

                c00 = __builtin_amdgcn_wmma_f32_16x16x4_f32(
                          false, a0[ks], false, b0, (short)0, c00, false, false);
                c01 = __builtin_amdgcn_wmma_f32_16x16x4_f32(
                          false, a0[ks], false, b1, (short)0, c01, false, false);
                c10 = __builtin_amdgcn_wmma_f32_16x16x4_f32(
                          false, a1[ks], false, b0, (short)0, c10, false, false);
                c11 = __builtin_amdgcn_wmma_f32_16x16x4_f32(
                          false, a1[ks], false, b1, (short)0, c11, false, false);
            }
            // C/D layout: VGPR r -> row r (lanes 0-15) / r+8 (lanes 16-31); col = lane%16
#pragma unroll
            for (int r = 0; r < 8; ++r) {
                const int row = 8 * half + r;
                os[row * 32 + l16]             = c00[r];
                os[row * 32 + 16 + l16]        = c01[r];
                os[(16 + row) * 32 + l16]      = c10[r];
                os[(16 + row) * 32 + 16 + l16] = c11[r];
            }
        } else {
            // Padded region: write exact zeros (d_out is poisoned by harness).
#pragma unroll
            for (int r = 0; r < 8; ++r) {
                const int row = 8 * half + r;
                os[row * 32 + l16]             = 0.0f;
                os[row * 32 + 16 + l16]        = 0.0f;
                os[(16 + row) * 32 + l16]      = 0.0f;
                os[(16 + row) * 32 + 16 + l16] = 0.0f;
            }
        }
    }
}

extern "C" void kernel_launch(void* const* d_in, const int* in_sizes, int n_in,
                              void* d_out, int out_size, void* d_ws, size_t ws_size,
                              hipStream_t stream) {
    (void)in_sizes; (void)n_in; (void)out_size; (void)ws_size;
    const float* x = (const float*)d_in[0];
    float* out = (float*)d_out;

    double* wsd = (double*)d_ws;                            // 15360 doubles
    float*  Qf  = (float*)((char*)d_ws + 15360 * sizeof(double));

    spool_setup_kernel<<<1, 256, 0, stream>>>(wsd, Qf);
    spool_main_kernel<<<256, 256, 0, stream>>>(x, Qf, out);
}